// TrainingMatchedGraphSageEncoder_33122787786831
// MI455X (gfx1250) — compile-verified
//
#include <hip/hip_runtime.h>
#include <stdint.h>

typedef __attribute__((ext_vector_type(2))) float v2f;
typedef __attribute__((ext_vector_type(8))) float v8f;
typedef __attribute__((ext_vector_type(4))) int   v4i;

static constexpr int kNodes = 50000;
static constexpr int kEdges = 800000;
static constexpr int kDIn   = 128;
static constexpr int kDHid  = 256;
static constexpr int kDOut  = 128;

// CDNA5 async global->LDS staging (ASYNCcnt path), guarded: falls back to
// plain LDS stores if the toolchain doesn't declare the builtins.
#if defined(__has_builtin)
#  if __has_builtin(__builtin_amdgcn_global_load_async_to_lds_b128) && \
      __has_builtin(__builtin_amdgcn_s_wait_asynccnt)
#    define USE_ASYNC_LDS 1
#  endif
#endif

#ifdef USE_ASYNC_LDS
typedef v4i __attribute__((address_space(1)))* gv4i_t;   // global int4*
typedef v4i __attribute__((address_space(3)))* lv4i_t;   // LDS int4*
#endif

__device__ __forceinline__ void stage16(const float* __restrict__ g, float* l) {
#ifdef USE_ASYNC_LDS
  // generic->AS(1)/AS(3) via integer round-trip: on gfx1250 the generic LDS
  // address's low 32 bits are the LDS byte offset (flat aperture rules).
  gv4i_t gp = (gv4i_t)(unsigned long long)(uintptr_t)g;
  lv4i_t lp = (lv4i_t)(unsigned int)(uintptr_t)l;
  __builtin_amdgcn_global_load_async_to_lds_b128(gp, lp, 0, 0);
#else
  *(float4*)l = *(const float4*)g;
#endif
}

__device__ __forceinline__ void stage_fence() {
#ifdef USE_ASYNC_LDS
  __builtin_amdgcn_s_wait_asynccnt(0);
#endif
}

__global__ void zero_f32_kernel(float* __restrict__ p, long long n) {
  long long i = (long long)blockIdx.x * blockDim.x + threadIdx.x;
  long long stride = (long long)gridDim.x * blockDim.x;
  for (; i < n; i += stride) p[i] = 0.0f;
}

__global__ void degree_kernel(const long long* __restrict__ ei,
                              float* __restrict__ deg, int E) {
  int e = blockIdx.x * blockDim.x + threadIdx.x;
  if (e < E) {
    int dst = (int)ei[E + e];
    atomicAdd(deg + dst, 1.0f);
  }
}

__global__ void rdeg_kernel(float* __restrict__ deg, int n) {
  int i = blockIdx.x * blockDim.x + threadIdx.x;
  if (i < n) deg[i] = 1.0f / fmaxf(deg[i], 1.0f);
}

// Segment-MEAN fused into the scatter: each contribution is pre-scaled by
// rdeg[dst] (graph-static), so agg already holds mean_j x_j after this pass.
__global__ void scatter_add_kernel(const float* __restrict__ x,
                                   const long long* __restrict__ ei,
                                   const float* __restrict__ rdeg,
                                   float* __restrict__ agg, int E, int K) {
  const int qpe = K >> 2;                       // float4 groups per edge
  int tid = blockIdx.x * blockDim.x + threadIdx.x;
  if (tid >= E * qpe) return;
  int e = tid / qpe;
  int q = (tid - e * qpe) << 2;
  int src = (int)ei[e];
  int dst = (int)ei[E + e];
  const float4 v = *(const float4*)(x + (size_t)src * K + q);
  const float s = rdeg[dst];
  float* o = agg + (size_t)dst * K + q;
  atomicAdd(o + 0, v.x * s);
  atomicAdd(o + 1, v.y * s);
  atomicAdd(o + 2, v.z * s);
  atomicAdd(o + 3, v.w * s);
}

__global__ void mat_add_kernel(const float* __restrict__ a,
                               const float* __restrict__ b,
                               float* __restrict__ o, int n) {
  int i = blockIdx.x * blockDim.x + threadIdx.x;
  if (i < n) o[i] = a[i] + b[i];
}

__global__ void mat_add_identity_kernel(const float* __restrict__ a,
                                        float* __restrict__ o, int dim) {
  int i = blockIdx.x * blockDim.x + threadIdx.x;
  if (i < dim * dim) o[i] = a[i] + (((i / dim) == (i % dim)) ? 1.0f : 0.0f);
}

// Fused SAGE layer GEMM with LDS block tiling:
//   out[n,m] = act( agg[n,:] @ Wl^T + X[n,:] @ Wr^T + b[m] )
// Block = 256 threads = 8 wave32s arranged 2 (rows) x 4 (cols);
// block tile = 32 rows x 64 cols, K-chunked by 64 through LDS.
// Each wave computes one 16x16 tile with V_WMMA_F32_16X16X4_F32:
//   A-frag: lane = kg*16+m holds A[m, k+kg*2 .. +1]  (8B ds_load)
//   B-frag: lane = kg*16+n holds B[k+kg*2 .. +1, n] = W[n, k..]
//   C/D:    VGPR r, lane -> row = r + (lane>=16)*8, col = lane&15
static constexpr int KC  = 64;
static constexpr int KCP = KC + 4;   // pad: 16B-aligned rows, conflict-free b64 reads

__global__ void __launch_bounds__(256)
sage_gemm_lds_kernel(const float* __restrict__ agg,
                     const float* __restrict__ Wl, const float* __restrict__ X,
                     const float* __restrict__ Wr, const float* __restrict__ bias,
                     float* __restrict__ out, int N, int K, int M, int do_relu) {
  __shared__ __align__(16) float sA[2][32 * KCP];   // [0]=agg (pre-meaned), [1]=x
  __shared__ __align__(16) float sW[2][64 * KCP];   // [0]=Wl cols, [1]=Wr cols

  const int tid  = threadIdx.x;
  const int lane = tid & 31;
  const int wid  = tid >> 5;          // 0..7
  const int rowTile = wid >> 2;       // 0..1
  const int colTile = wid & 3;        // 0..3
  const int m16 = lane & 15;
  const int kg  = lane >> 4;          // K pair select

  const int colBlocks = M >> 6;
  const int blockRow = blockIdx.x / colBlocks;
  const int blockCol = blockIdx.x % colBlocks;
  const int row0 = blockRow * 32;
  const int col0 = blockCol * 64;

  const int  tileRow  = row0 + rowTile * 16;
  const bool rowValid = (tileRow < N);            // wave-uniform

  v8f acc = {};

  for (int kc = 0; kc < K; kc += KC) {
    __syncthreads();                               // LDS reuse fence
    // --- stage A block: 2 matrices x 32 rows x KC, float4 granularity ---
    for (int i = tid; i < (32 * KC) / 4; i += 256) {
      int r  = i / (KC / 4);
      int c4 = (i % (KC / 4)) * 4;
      int gr = row0 + r;
      float* la = &sA[0][r * KCP + c4];
      float* lx = &sA[1][r * KCP + c4];
      if (gr < N) {
        stage16(agg + (size_t)gr * K + kc + c4, la);
        stage16(X   + (size_t)gr * K + kc + c4, lx);
      } else {
        *(float4*)la = make_float4(0.f, 0.f, 0.f, 0.f);
        *(float4*)lx = make_float4(0.f, 0.f, 0.f, 0.f);
      }
    }
    // --- stage W block: 2 matrices x 64 cols x KC ---
    for (int i = tid; i < (64 * KC) / 4; i += 256) {
      int c  = i / (KC / 4);
      int c4 = (i % (KC / 4)) * 4;
      stage16(Wl + (size_t)(col0 + c) * K + kc + c4, &sW[0][c * KCP + c4]);
      stage16(Wr + (size_t)(col0 + c) * K + kc + c4, &sW[1][c * KCP + c4]);
    }
    stage_fence();                                 // ASYNCcnt -> 0 (async path)
    __syncthreads();

    if (rowValid) {
      const float* aP = &sA[0][(rowTile * 16 + m16) * KCP + (kg << 1)];
      const float* xP = &sA[1][(rowTile * 16 + m16) * KCP + (kg << 1)];
      const float* lP = &sW[0][(colTile * 16 + m16) * KCP + (kg << 1)];
      const float* rP = &sW[1][(colTile * 16 + m16) * KCP + (kg << 1)];
#pragma unroll
      for (int k = 0; k < KC; k += 4) {
        v2f a1 = *(const v2f*)(aP + k);
        v2f b1 = *(const v2f*)(lP + k);
        acc = __builtin_amdgcn_wmma_f32_16x16x4_f32(false, a1, false, b1,
                                                    (short)0, acc, false, false);
        v2f a2 = *(const v2f*)(xP + k);
        v2f b2 = *(const v2f*)(rP + k);
        acc = __builtin_amdgcn_wmma_f32_16x16x4_f32(false, a2, false, b2,
                                                    (short)0, acc, false, false);
      }
    }
  }

  if (rowValid) {
    const int ocol = col0 + colTile * 16 + m16;
    const float bv = bias[ocol];
    float* oP = out + (size_t)(tileRow + (kg << 3)) * M + ocol;
#pragma unroll
    for (int r = 0; r < 8; ++r) {
      float v = acc[r] + bv;
      if (do_relu) v = fmaxf(v, 0.0f);
      oP[(size_t)r * M] = v;
    }
  }
}

extern "C" void kernel_launch(void* const* d_in, const int* in_sizes, int n_in,
                              void* d_out, int out_size, void* d_ws, size_t ws_size,
                              hipStream_t stream) {
  const float*     x   = (const float*)d_in[0];
  const long long* ei  = (const long long*)d_in[1];
  const float*     Wl1 = (const float*)d_in[2];
  const float*     b1  = (const float*)d_in[3];
  const float*     Wr1 = (const float*)d_in[4];
  const float*     Wl2 = (const float*)d_in[5];
  const float*     b2  = (const float*)d_in[6];
  const float*     Wr2 = (const float*)d_in[7];
  const float*     Wl3 = (const float*)d_in[8];
  const float*     b3  = (const float*)d_in[9];
  const float*     Wr3 = (const float*)d_in[10];
  const float*     Wpr = (const float*)d_in[11];
  float* out = (float*)d_out;
  (void)in_sizes; (void)n_in; (void)out_size; (void)ws_size;

  char* ws = (char*)d_ws;
  size_t off = 0;
  auto take = [&](size_t bytes) -> float* {
    float* p = (float*)(ws + off);
    off = (off + bytes + 255) & ~(size_t)255;
    return p;
  };
  float* rdeg = take((size_t)kNodes * 4);
  float* agg  = take((size_t)kNodes * kDHid * 4);
  float* h1   = take((size_t)kNodes * kDHid * 4);
  float* h2   = take((size_t)kNodes * kDHid * 4);
  float* Wr2I = take((size_t)kDHid * kDHid * 4);   // W_r2 + I  (layer-2 residual fold)
  float* W3s  = take((size_t)kDOut * kDHid * 4);   // W_r3 + W_proj (layer-3 residual fold)

  const int rowBlocks = (kNodes + 31) / 32;        // 1563

  // --- degree -> reciprocal-mean scale (graph static across layers) ---
  zero_f32_kernel<<<1024, 256, 0, stream>>>(rdeg, kNodes);
  degree_kernel<<<(kEdges + 255) / 256, 256, 0, stream>>>(ei, rdeg, kEdges);
  rdeg_kernel<<<(kNodes + 255) / 256, 256, 0, stream>>>(rdeg, kNodes);

  // --- fold residuals into the "right" weights ---
  mat_add_identity_kernel<<<(kDHid * kDHid + 255) / 256, 256, 0, stream>>>(Wr2, Wr2I, kDHid);
  mat_add_kernel<<<(kDOut * kDHid + 255) / 256, 256, 0, stream>>>(Wr3, Wpr, W3s, kDOut * kDHid);

  // ---- layer 1: d_in=128 -> d_hid=256, ReLU ----
  zero_f32_kernel<<<2048, 256, 0, stream>>>(agg, (long long)kNodes * kDIn);
  scatter_add_kernel<<<(kEdges * (kDIn / 4) + 255) / 256, 256, 0, stream>>>(
      x, ei, rdeg, agg, kEdges, kDIn);
  sage_gemm_lds_kernel<<<rowBlocks * (kDHid / 64), 256, 0, stream>>>(
      agg, Wl1, x, Wr1, b1, h1, kNodes, kDIn, kDHid, 1);

  // ---- layer 2: d_hid -> d_hid, +residual (folded into Wr2+I), ReLU ----
  zero_f32_kernel<<<2048, 256, 0, stream>>>(agg, (long long)kNodes * kDHid);
  scatter_add_kernel<<<(kEdges * (kDHid / 4) + 255) / 256, 256, 0, stream>>>(
      h1, ei, rdeg, agg, kEdges, kDHid);
  sage_gemm_lds_kernel<<<rowBlocks * (kDHid / 64), 256, 0, stream>>>(
      agg, Wl2, h1, Wr2I, b2, h2, kNodes, kDHid, kDHid, 1);

  // ---- layer 3: d_hid -> d_out, + projected residual (folded), no ReLU ----
  zero_f32_kernel<<<2048, 256, 0, stream>>>(agg, (long long)kNodes * kDHid);
  scatter_add_kernel<<<(kEdges * (kDHid / 4) + 255) / 256, 256, 0, stream>>>(
      h2, ei, rdeg, agg, kEdges, kDHid);
  sage_gemm_lds_kernel<<<rowBlocks * (kDOut / 64), 256, 0, stream>>>(
      agg, Wl3, h2, W3s, b3, out, kNodes, kDHid, kDOut, 0);
}